// FNNModel_60576218742787
// MI455X (gfx1250) — compile-verified
//
#include <hip/hip_runtime.h>
#include <math.h>

// ---------------------------------------------------------------------------
// FNN model for MI455X (gfx1250, wave32).
//   * All GEMMs use V_WMMA_F32_16X16X4_F32 (CDNA5 native fp32 matrix op),
//     128x128 block tiles, 8 waves x (32x64 C region) = 2x4 fragments/wave.
//   * GEMM tiles staged to LDS via GLOBAL_LOAD_ASYNC_TO_LDS_B32 (ASYNCcnt)
//     when the builtin exists; plain LDS stores otherwise.
//   * Setup exploits Qmat==0, S structure, and A = T4 (x) I (hardcoded T4^-1).
//   * Recurrent scan: input projection hoisted to a WMMA GEMM; remaining
//     latency-bound part runs in one persistent 1024-thread workgroup.
// ws requirement: ~110 MB of float scratch.
// ---------------------------------------------------------------------------

#define NDIM   512
#define MDIM   512
#define QD     512
#define PDIM   768
#define LLAY   4
#define TSTEP  2048
#define LQ     2048
#define NN2    2560
#define EPSV   0.001f
#define PITERS 24

typedef float v2f __attribute__((ext_vector_type(2)));
typedef float v8f __attribute__((ext_vector_type(8)));

#if __has_builtin(__builtin_amdgcn_global_load_async_to_lds_b32) && \
    __has_builtin(__builtin_amdgcn_s_wait_asynccnt)
#define HAVE_ASYNC_LDS 1
#define AS_GLOBAL __attribute__((address_space(1)))
#define AS_SHARED __attribute__((address_space(3)))
#else
#define HAVE_ASYNC_LDS 0
#endif

// scal[] slots
#define S_NORM   0
#define S_INV    1
#define S_DELTA  2
#define S_LAMBDA 3

// T4 = tridiag(1, -0.5) for L=4, gamma=0.5;  A^-1 = T4^-1 (x) I_512
__device__ __constant__ float T4INV[16] = {
    1.6f, 1.2f, 0.8f, 0.4f,
    1.2f, 2.4f, 1.6f, 0.8f,
    0.8f, 1.6f, 2.4f, 1.2f,
    0.4f, 0.8f, 1.2f, 1.6f };

// ---------------------------------------------------------------------------
// Tiled fp32 WMMA GEMM:  C(MxN) = alpha * op(A)(MxK) * op(B)(KxN) + beta * C
// TA: A stored (K x M) row-major, op(A)=A^T.  TB: B stored (N x K), op(B)=B^T.
// Requirements: M multiple of 128, N multiple of 128, K multiple of 32
// (true for every call below).  Block = 256 threads = 8 waves (wave32),
// waves arranged 4 (M) x 2 (N); each wave owns 32x64 of C (2x4 fragments,
// 8 v_wmma per K-microstep -> 0.75 LDS fragment loads per WMMA).
// ---------------------------------------------------------------------------
template <bool TA, bool TB>
__global__ __launch_bounds__(256) void gemm_f32_wmma(
    const float* __restrict__ A, const float* __restrict__ B,
    float* __restrict__ C, int M, int N, int K,
    int lda, int ldb, int ldc, float alpha, float beta)
{
    (void)M; (void)N;
    __shared__ float Asm[128][33];   // [m][k] (+1 pad vs 64-bank LDS)
    __shared__ float Bsm[32][129];   // [k][n]

    const int tid   = threadIdx.x;
    const int lane  = tid & 31;
    const int wave  = tid >> 5;
    const int waveM = wave >> 1;      // 0..3
    const int waveN = wave & 1;       // 0..1
    const int lo    = lane & 15;
    const int hi    = lane >> 4;      // selects K pair / M-half per ISA layout
    const int m0    = blockIdx.y * 128;
    const int n0    = blockIdx.x * 128;
    const int rA    = waveM * 32;
    const int cB    = waveN * 64;

    v8f acc[2][4] = {};

    for (int kc = 0; kc < K; kc += 32) {
        // ---- stage tiles in LDS (async-to-LDS path when available) -------
        for (int idx = tid; idx < 128 * 32; idx += 256) {
            int r = idx >> 5, c = idx & 31;
            const float* ga = TA ? &A[(size_t)(kc + c) * lda + (m0 + r)]
                                 : &A[(size_t)(m0 + r) * lda + (kc + c)];
#if HAVE_ASYNC_LDS
            __builtin_amdgcn_global_load_async_to_lds_b32(
                (AS_GLOBAL int*)(void*)const_cast<float*>(ga),
                (AS_SHARED int*)(void*)&Asm[r][c], 0, 0);
#else
            Asm[r][c] = *ga;
#endif
        }
        for (int idx = tid; idx < 32 * 128; idx += 256) {
            int r = idx >> 7, c = idx & 127;
            const float* gb = TB ? &B[(size_t)(n0 + c) * ldb + (kc + r)]
                                 : &B[(size_t)(kc + r) * ldb + (n0 + c)];
#if HAVE_ASYNC_LDS
            __builtin_amdgcn_global_load_async_to_lds_b32(
                (AS_GLOBAL int*)(void*)const_cast<float*>(gb),
                (AS_SHARED int*)(void*)&Bsm[r][c], 0, 0);
#else
            Bsm[r][c] = *gb;
#endif
        }
        // prefetch the tile after this one (global_prefetch_b8 hint)
        if (kc + 32 < K) {
            int r = tid >> 5, c = tid & 31;
            const float* pa = TA ? &A[(size_t)(kc + 32 + c) * lda + (m0 + r)]
                                 : &A[(size_t)(m0 + r) * lda + (kc + 32 + c)];
            __builtin_prefetch(pa, 0, 1);
            int rb = tid >> 7, cb = tid & 127;
            const float* pb = TB ? &B[(size_t)(n0 + cb) * ldb + (kc + 32 + rb)]
                                 : &B[(size_t)(kc + 32 + rb) * ldb + (n0 + cb)];
            __builtin_prefetch(pb, 0, 1);
        }
#if HAVE_ASYNC_LDS
        __builtin_amdgcn_s_wait_asynccnt(0);
#endif
        __syncthreads();

#pragma unroll
        for (int kk = 0; kk < 8; ++kk) {
            // ISA 16x16x4 f32 layouts:
            //  A 16x4: lane m=lo, v[r] = A[m][kk*4 + r + 2*hi]
            //  B 4x16: lane n=lo, v[r] = B[kk*4 + r + 2*hi][n]
            const int kb = kk * 4 + 2 * hi;
            v2f a[2], b[4];
#pragma unroll
            for (int ms = 0; ms < 2; ++ms) {
                a[ms].x = Asm[rA + 16 * ms + lo][kb];
                a[ms].y = Asm[rA + 16 * ms + lo][kb + 1];
            }
#pragma unroll
            for (int ns = 0; ns < 4; ++ns) {
                b[ns].x = Bsm[kb][cB + 16 * ns + lo];
                b[ns].y = Bsm[kb + 1][cB + 16 * ns + lo];
            }
#pragma unroll
            for (int ms = 0; ms < 2; ++ms)
#pragma unroll
                for (int ns = 0; ns < 4; ++ns)
                    acc[ms][ns] = __builtin_amdgcn_wmma_f32_16x16x4_f32(
                        false, a[ms], false, b[ns], (short)0, acc[ms][ns],
                        false, false);
        }
        __syncthreads();
    }

    // ---- writeback (C frag: VGPR r -> row r + 8*hi, col = lo) -------------
#pragma unroll
    for (int ms = 0; ms < 2; ++ms)
#pragma unroll
        for (int ns = 0; ns < 4; ++ns) {
            const int col = n0 + cB + 16 * ns + lo;
#pragma unroll
            for (int r = 0; r < 8; ++r) {
                int row = m0 + rA + 16 * ms + r + 8 * hi;
                size_t i0 = (size_t)row * ldc + col;
                C[i0] = alpha * acc[ms][ns][r] + beta * C[i0];
            }
        }
}

// ---------------------------------------------------------------------------
// Gauss-Jordan inverse of a 512x512 diagonally-dominant matrix.
// Single workgroup, 1024 threads (32 wave32 waves), Aug is 512x1024 [A | I].
// ---------------------------------------------------------------------------
__global__ __launch_bounds__(256) void k_init_aug(const float* __restrict__ src,
                                                  float* __restrict__ Aug)
{
    int idx = blockIdx.x * 256 + threadIdx.x;
    if (idx >= 512 * 1024) return;
    int i = idx >> 10, j = idx & 1023;
    Aug[idx] = (j < 512) ? src[i * 512 + j] : ((j - 512 == i) ? 1.0f : 0.0f);
}

__global__ __launch_bounds__(1024) void k_gauss_jordan(float* __restrict__ Aug)
{
    __shared__ float fac[512];
    __shared__ float pivinv;
    const int tid = threadIdx.x;            // 0..1023 == column id
    for (int k = 0; k < 512; ++k) {
        if (tid == 0) pivinv = 1.0f / Aug[(size_t)k * 1024 + k];
        __syncthreads();
        Aug[(size_t)k * 1024 + tid] *= pivinv;     // normalize pivot row
        __syncthreads();
        if (tid < 512) fac[tid] = (tid == k) ? 0.0f : Aug[(size_t)tid * 1024 + k];
        __syncthreads();
        const float pivval = Aug[(size_t)k * 1024 + tid];
        for (int r = 0; r < 512; ++r)
            if (r != k) Aug[(size_t)r * 1024 + tid] -= fac[r] * pivval;
        __syncthreads();
    }
}

__global__ __launch_bounds__(256) void k_copy_inv(const float* __restrict__ Aug,
                                                  float* __restrict__ dst)
{
    int idx = blockIdx.x * 256 + threadIdx.x;
    if (idx >= 512 * 512) return;
    int i = idx >> 9, j = idx & 511;
    dst[idx] = Aug[(size_t)i * 1024 + 512 + j];
}

// ---------------------------------------------------------------------------
// Power-iteration building blocks (spectral norms; fixed iterations).
// ---------------------------------------------------------------------------
__global__ __launch_bounds__(256) void k_setconst(float* v, int n, float val)
{
    int i = blockIdx.x * 256 + threadIdx.x;
    if (i < n) v[i] = val;
}
__global__ __launch_bounds__(256) void k_mv(const float* __restrict__ A,
                                            const float* __restrict__ v,
                                            float* __restrict__ y,
                                            int rows, int cols, int lda)
{
    int i = blockIdx.x * 256 + threadIdx.x;
    if (i >= rows) return;
    const float* a = A + (size_t)i * lda;
    float s = 0.0f;
    for (int j = 0; j < cols; ++j) s = fmaf(a[j], v[j], s);
    y[i] = s;
}
__global__ __launch_bounds__(256) void k_mvT(const float* __restrict__ A,
                                             const float* __restrict__ v,
                                             float* __restrict__ y,
                                             int rows, int cols, int lda)
{
    int j = blockIdx.x * 256 + threadIdx.x;
    if (j >= cols) return;
    float s = 0.0f;
    for (int i = 0; i < rows; ++i) s = fmaf(A[(size_t)i * lda + j], v[i], s);
    y[j] = s;
}
__global__ __launch_bounds__(1024) void k_norm(const float* __restrict__ y,
                                               int n, float* scal)
{
    __shared__ float red[1024];
    float s = 0.0f;
    for (int i = threadIdx.x; i < n; i += 1024) s = fmaf(y[i], y[i], s);
    red[threadIdx.x] = s;
    __syncthreads();
    for (int st = 512; st > 0; st >>= 1) {
        if (threadIdx.x < st) red[threadIdx.x] += red[threadIdx.x + st];
        __syncthreads();
    }
    if (threadIdx.x == 0) {
        float nn = sqrtf(red[0]);
        scal[S_NORM] = nn;
        scal[S_INV]  = (nn > 1e-30f) ? 1.0f / nn : 0.0f;
    }
}
__global__ __launch_bounds__(256) void k_scale(const float* y, float* v, int n,
                                               const float* scal)
{
    int i = blockIdx.x * 256 + threadIdx.x;
    if (i < n) v[i] = y[i] * scal[S_INV];
}
__global__ void k_store_sigma(float* scal, int dst)
{
    if (threadIdx.x == 0) scal[dst] = scal[S_NORM] + EPSV;
}

// ---------------------------------------------------------------------------
// Matrix-assembly elementwise kernels.
// ---------------------------------------------------------------------------
// G = X3^T X3 + Z3^T Z3 (from GEMMs);  Mmat = G + (Y3 - Y3^T) + eps I
__global__ __launch_bounds__(256) void k_build_IpM(const float* __restrict__ G,
                                                   const float* __restrict__ Y3,
                                                   float* __restrict__ IplusM,
                                                   float* __restrict__ ImM)
{
    int idx = blockIdx.x * 256 + threadIdx.x;
    if (idx >= 512 * 512) return;
    int i = idx >> 9, j = idx & 511;
    float m = G[idx] + Y3[i * 512 + j] - Y3[j * 512 + i] + ((i == j) ? EPSV : 0.0f);
    float d = (i == j) ? 1.0f : 0.0f;
    IplusM[idx] = d + m;
    ImM[idx]    = d - m;
}
// D22 (P x M) = S^T + 0.5 * Nmat ; Nmat top = Ntop, bottom = -2 * (Z3 @ Ainv)
__global__ __launch_bounds__(256) void k_build_D22(const float* __restrict__ Ntop,
                                                   const float* __restrict__ NbotR,
                                                   float* __restrict__ D22)
{
    int idx = blockIdx.x * 256 + threadIdx.x;
    if (idx >= PDIM * MDIM) return;
    int p = idx / MDIM, m = idx % MDIM;
    float nv = (p < 512) ? Ntop[p * 512 + m] : (-2.0f * NbotR[(p - 512) * 512 + m]);
    D22[idx] = (((p - 256) == m) ? 0.5f : 0.0f) + 0.5f * nv;
}
// R1 = S@D22 + (S@D22)^T + eps I ;  (S@D22)[i][j] = 0.5 * D22[256+i][j]
__global__ __launch_bounds__(256) void k_build_R1(const float* __restrict__ D22,
                                                  float* __restrict__ R1)
{
    int idx = blockIdx.x * 256 + threadIdx.x;
    if (idx >= 512 * 512) return;
    int i = idx >> 9, j = idx & 511;
    R1[idx] = 0.5f * D22[(256 + i) * 512 + j] + 0.5f * D22[(256 + j) * 512 + i]
            + ((i == j) ? EPSV : 0.0f);
}
// Wmat (NN x M): rows<512: 0.5*C[256+m][i]; rows 512..1023: -0.5*Wu; else 0
__global__ __launch_bounds__(256) void k_build_Wmat(const float* __restrict__ Cm,
                                                    const float* __restrict__ Wu,
                                                    float* __restrict__ Wmat)
{
    int idx = blockIdx.x * 256 + threadIdx.x;
    if (idx >= NN2 * MDIM) return;
    int i = idx / MDIM, m = idx % MDIM;
    float v;
    if (i < 512)       v = 0.5f * Cm[(256 + m) * NDIM + i];
    else if (i < 1024) v = -0.5f * Wu[(i - 512) * MDIM + m];
    else               v = 0.0f;
    Wmat[idx] = v;
}
__global__ __launch_bounds__(256) void k_H_add_eps(float* __restrict__ H)
{
    int i = blockIdx.x * 256 + threadIdx.x;
    if (i < NN2) H[(size_t)i * NN2 + i] += EPSV;
}
// H21_star (LQ x N) and Diff = H21_star - H21
__global__ __launch_bounds__(256) void k_build_H21s(const float* __restrict__ H,
                                                    const float* __restrict__ W1,
                                                    float* __restrict__ H21s,
                                                    float* __restrict__ Diff)
{
    int idx = blockIdx.x * 256 + threadIdx.x;
    if (idx >= LQ * NDIM) return;
    int i = idx / NDIM, j = idx % NDIM;
    float h21 = H[(size_t)j * NN2 + 512 + i];     // H21[i][j] = H[j][N+i]
    if (i < QD) { H21s[idx] = h21 + W1[i * NDIM + j]; Diff[idx] = W1[i * NDIM + j]; }
    else        { H21s[idx] = 0.0f;                   Diff[idx] = -h21; }
}
// Pmat = H11 + delta I   (Qmat == 0)
__global__ __launch_bounds__(256) void k_build_Pmat(const float* __restrict__ H,
                                                    const float* __restrict__ scal,
                                                    float* __restrict__ Pm)
{
    int idx = blockIdx.x * 256 + threadIdx.x;
    if (idx >= NDIM * NDIM) return;
    int i = idx >> 9, j = idx & 511;
    Pm[idx] = H[(size_t)i * NN2 + j] + ((i == j) ? scal[S_DELTA] : 0.0f);
}
// P1 = H22 + delta I, with B^T Pmat B added into the last diagonal block
__global__ __launch_bounds__(256) void k_build_P1(const float* __restrict__ H,
                                                  const float* __restrict__ BtPB,
                                                  const float* __restrict__ scal,
                                                  float* __restrict__ P1)
{
    int idx = blockIdx.x * 256 + threadIdx.x;
    if (idx >= LQ * LQ) return;
    int i = idx / LQ, j = idx % LQ;
    float v = H[(size_t)(512 + i) * NN2 + 512 + j] + ((i == j) ? scal[S_DELTA] : 0.0f);
    if (i >= 1536 && j >= 1536) v += BtPB[(i - 1536) * QD + (j - 1536)];
    P1[idx] = v;
}
// LambdaW2 blocks b=0..2: 0.5*(P1[(b+1)Q+r][bQ+c] + W1_sub[b][r][c])
__global__ __launch_bounds__(256) void k_build_LW2(const float* __restrict__ P1,
                                                   const float* __restrict__ W1s,
                                                   float* __restrict__ LW2)
{
    int idx = blockIdx.x * 256 + threadIdx.x;
    if (idx >= 3 * QD * QD) return;
    int b = idx / (QD * QD), rem = idx % (QD * QD);
    int r = rem / QD, c = rem % QD;
    LW2[idx] = 0.5f * (P1[(size_t)((b + 1) * QD + r) * LQ + b * QD + c] + W1s[idx]);
}
// residual = P1 + LambdaW2 + LambdaW2^T  (in place; disjoint blocks)
__global__ __launch_bounds__(256) void k_residual(float* __restrict__ P1,
                                                  const float* __restrict__ LW2)
{
    int idx = blockIdx.x * 256 + threadIdx.x;
    if (idx >= 3 * QD * QD) return;
    int b = idx / (QD * QD), rem = idx % (QD * QD);
    int r = rem / QD, c = rem % QD;
    float v = LW2[idx];
    P1[(size_t)((b + 1) * QD + r) * LQ + b * QD + c] += v;
    P1[(size_t)(b * QD + c) * LQ + (b + 1) * QD + r] += v;
}
// Ms = (T4^-1 (x) I) @ residual
__global__ __launch_bounds__(256) void k_msolve(const float* __restrict__ P1,
                                                float* __restrict__ Ms)
{
    int idx = blockIdx.x * 256 + threadIdx.x;
    if (idx >= LQ * LQ) return;
    int i = idx / LQ, j = idx % LQ;
    int bi = i >> 9, r = i & 511;
    float s = 0.0f;
#pragma unroll
    for (int b = 0; b < 4; ++b)
        s = fmaf(T4INV[bi * 4 + b], P1[(size_t)((b << 9) + r) * LQ + j], s);
    Ms[idx] = s;
}
// alpha_diag = lambda + softplus(param) + eps ;  inva = 1/alpha
__global__ __launch_bounds__(256) void k_alpha(const float* __restrict__ adp,
                                               const float* __restrict__ scal,
                                               float* __restrict__ inva)
{
    int i = blockIdx.x * 256 + threadIdx.x;
    if (i >= LQ) return;
    float sp = log1pf(expf(adp[i]));
    inva[i] = 1.0f / (scal[S_LAMBDA] + sp + EPSV);
}
// Wu_eff0 = inva[:Q] * Wu  (1/gamma * inva * 0.5*Wu == inva*Wu)
__global__ __launch_bounds__(256) void k_wue(const float* __restrict__ Wu,
                                             const float* __restrict__ inva,
                                             float* __restrict__ WuE)
{
    int idx = blockIdx.x * 256 + threadIdx.x;
    if (idx >= QD * MDIM) return;
    WuE[idx] = inva[idx / MDIM] * Wu[idx];
}
// W1_eff = -2 * inva * H21_star   (in place on H21s)
__global__ __launch_bounds__(256) void k_w1e(float* __restrict__ H21s,
                                             const float* __restrict__ inva)
{
    int idx = blockIdx.x * 256 + threadIdx.x;
    if (idx >= LQ * NDIM) return;
    H21s[idx] *= (-2.0f * inva[idx / NDIM]);
}
// W2_eff[b] = 2 * inva[(b+1)Q + r] * LW2[b]
__global__ __launch_bounds__(256) void k_w2e(const float* __restrict__ LW2,
                                             const float* __restrict__ inva,
                                             float* __restrict__ W2E)
{
    int idx = blockIdx.x * 256 + threadIdx.x;
    if (idx >= 3 * QD * QD) return;
    int b = idx / (QD * QD), r = (idx / QD) % QD;
    W2E[idx] = 2.0f * inva[(b + 1) * QD + r] * LW2[idx];
}
__global__ __launch_bounds__(256) void k_biasY(float* __restrict__ Yo,
                                               const float* __restrict__ by)
{
    int idx = blockIdx.x * 256 + threadIdx.x;
    if (idx < TSTEP * PDIM) Yo[idx] += by[idx % PDIM];
}

// ---------------------------------------------------------------------------
// Latency-bound recurrent scan. One persistent workgroup, 1024 threads
// (32 wave32 waves); x/prev live in LDS; weights (8 MB total) stream from L2.
// 2 threads per output row, each a 256-wide partial dot.
// ---------------------------------------------------------------------------
__global__ __launch_bounds__(1024) void k_recurrence(
    const float* __restrict__ Upre,  // T x Q : u @ Wu_eff0^T
    const float* __restrict__ W1E,   // L x Q x N
    const float* __restrict__ W2E,   // 3 x Q x Q (layers 1..3)
    const float* __restrict__ Bm,    // N x Q
    const float* __restrict__ bz,    // L x Q
    const float* __restrict__ bx,    // N
    const float* __restrict__ x0,    // N
    float* __restrict__ Xtm1,        // T x N (ws)
    float* __restrict__ hout)        // T x N (d_out)
{
    __shared__ float x[NDIM], prev[QD], nxt[QD];
    __shared__ float part[1024];
    const int tid  = threadIdx.x;
    const int row  = tid >> 1;
    const int half = tid & 1;
    const int j0   = half * 256;

    if (tid < NDIM) x[tid] = x0[tid];
    __syncthreads();

    for (int t = 0; t < TSTEP; ++t) {
        // ---- layer 0: prev = tanh(Upre[t] + W1E[0] @ x + bz[0]) ----------
        {
            const float* w1 = W1E + (size_t)row * NDIM;
            float s = 0.0f;
            for (int j = 0; j < 256; ++j) s = fmaf(w1[j0 + j], x[j0 + j], s);
            part[tid] = s;
        }
        __syncthreads();
        if (!half) {
            float v = part[tid] + part[tid + 1] + Upre[(size_t)t * QD + row] + bz[row];
            prev[row] = tanhf(v);
        }
        __syncthreads();
        // ---- layers 1..3 -------------------------------------------------
        for (int l = 1; l < LLAY; ++l) {
            const float* w1 = W1E + (size_t)l * QD * NDIM + (size_t)row * NDIM;
            const float* w2 = W2E + (size_t)(l - 1) * QD * QD + (size_t)row * QD;
            float s = 0.0f;
            for (int j = 0; j < 256; ++j) {
                int jj = j0 + j;
                s = fmaf(w1[jj], x[jj], s);
                s = fmaf(w2[jj], prev[jj], s);
            }
            part[tid] = s;
            __syncthreads();
            if (!half) {
                float v = part[tid] + part[tid + 1] + prev[row] + bz[l * QD + row];
                nxt[row] = tanhf(v);
            }
            __syncthreads();
            if (!half) prev[row] = nxt[row];
            __syncthreads();
        }
        // ---- x_next = B @ prev + bx; record outputs ---------------------
        {
            const float* br = Bm + (size_t)row * QD;
            float s = 0.0f;
            for (int j = 0; j < 256; ++j) s = fmaf(br[j0 + j], prev[j0 + j], s);
            part[tid] = s;
        }
        __syncthreads();
        if (!half) {
            float xn = part[tid] + part[tid + 1] + bx[row];
            Xtm1[(size_t)t * NDIM + row] = x[row];
            hout[(size_t)t * NDIM + row] = xn;
            nxt[row] = xn;
        }
        __syncthreads();
        if (!half) x[row] = nxt[row];
        __syncthreads();
    }
}

// ---------------------------------------------------------------------------
// Host orchestration (all on `stream`; graph-capture safe: no malloc/sync).
// ---------------------------------------------------------------------------
static inline dim3 g1(int n) { return dim3((unsigned)((n + 255) / 256)); }

extern "C" void kernel_launch(void* const* d_in, const int* in_sizes, int n_in,
                              void* d_out, int out_size, void* d_ws, size_t ws_size,
                              hipStream_t stream)
{
    (void)in_sizes; (void)n_in; (void)out_size; (void)ws_size;

    const float* u   = (const float*)d_in[0];    // T x M
    const float* x0  = (const float*)d_in[1];    // N
    const float* Cm  = (const float*)d_in[2];    // P x N
    const float* Ym  = (const float*)d_in[3];    // NN x NN
    const float* Wu  = (const float*)d_in[4];    // Q x M
    const float* W1  = (const float*)d_in[5];    // Q x N
    const float* X3  = (const float*)d_in[6];    // M x M
    const float* Y3  = (const float*)d_in[7];    // M x M
    const float* Z3  = (const float*)d_in[8];    // (P-M) x M
    const float* bx  = (const float*)d_in[9];    // M
    const float* by  = (const float*)d_in[10];   // P
    const float* bz  = (const float*)d_in[11];   // L x Q
    const float* W1s = (const float*)d_in[12];   // (L-1) x Q x Q
    const float* Bm  = (const float*)d_in[13];   // N x Q
    const float* adp = (const float*)d_in[14];   // L*N

    float* Yout = (float*)d_out;                         // T x P
    float* Pm   = Yout + (size_t)TSTEP * PDIM;           // N x N
    float* hout = Pm + (size_t)NDIM * NDIM;              // T x N

    // ---- workspace layout (floats) ---------------------------------------
    float* w = (float*)d_ws;
    float* G      = w; w += 512 * 512;
    float* IplusM = w; w += 512 * 512;
    float* ImM    = w; w += 512 * 512;
    float* Aug    = w; w += 512 * 1024;
    float* Ainv   = w; w += 512 * 512;
    float* Ntop   = w; w += 512 * 512;
    float* NbotR  = w; w += 256 * 512;
    float* D22    = w; w += PDIM * MDIM;
    float* R1     = w; w += 512 * 512;
    float* R1inv  = w; w += 512 * 512;
    float* Wmat   = w; w += (size_t)NN2 * MDIM;
    float* T1     = w; w += (size_t)NN2 * MDIM;
    float* H      = w; w += (size_t)NN2 * NN2;
    float* H21s   = w; w += (size_t)LQ * NDIM;          // later becomes W1_eff
    float* Diff   = w; w += (size_t)LQ * NDIM;
    float* BtP    = w; w += 512 * 512;
    float* BtPB   = w; w += 512 * 512;
    float* P1     = w; w += (size_t)LQ * LQ;            // later becomes residual
    float* Ms     = w; w += (size_t)LQ * LQ;
    float* LW2    = w; w += 3 * QD * QD;
    float* WuE    = w; w += QD * MDIM;
    float* W2E    = w; w += 3 * QD * QD;
    float* Upre   = w; w += (size_t)TSTEP * QD;
    float* Xtm1   = w; w += (size_t)TSTEP * NDIM;
    float* pv_v   = w; w += LQ;
    float* pv_w   = w; w += LQ;
    float* pv_t   = w; w += LQ;
    float* inva   = w; w += LQ;
    float* scal   = w; w += 16;

    const dim3 blk(256);

    // helper: invert 512x512 via Gauss-Jordan
    auto invert = [&](const float* src, float* dst) {
        k_init_aug<<<g1(512 * 1024), blk, 0, stream>>>(src, Aug);
        k_gauss_jordan<<<1, 1024, 0, stream>>>(Aug);
        k_copy_inv<<<g1(512 * 512), blk, 0, stream>>>(Aug, dst);
    };
    // helper: spectral norm of A (rows x cols) -> scal[dstIdx] = sigma + eps
    auto specnorm = [&](const float* A, int rows, int cols, int lda, int dstIdx) {
        k_setconst<<<g1(cols), blk, 0, stream>>>(pv_v, cols, 1.0f);
        for (int it = 0; it < PITERS; ++it) {
            k_mv <<<g1(rows), blk, 0, stream>>>(A, pv_v, pv_w, rows, cols, lda);
            k_mvT<<<g1(cols), blk, 0, stream>>>(A, pv_w, pv_t, rows, cols, lda);
            k_norm<<<1, 1024, 0, stream>>>(pv_t, cols, scal);
            k_scale<<<g1(cols), blk, 0, stream>>>(pv_t, pv_v, cols, scal);
        }
        k_mv<<<g1(rows), blk, 0, stream>>>(A, pv_v, pv_w, rows, cols, lda);
        k_norm<<<1, 1024, 0, stream>>>(pv_w, rows, scal);
        k_store_sigma<<<1, 1, 0, stream>>>(scal, dstIdx);
    };

    // ==== Setup ==========================================================
    // Mmat pieces: G = X3^T X3 + Z3^T Z3
    gemm_f32_wmma<true,  false><<<dim3(4, 4), blk, 0, stream>>>(
        X3, X3, G, 512, 512, 512, 512, 512, 512, 1.0f, 0.0f);
    gemm_f32_wmma<true,  false><<<dim3(4, 4), blk, 0, stream>>>(
        Z3, Z3, G, 512, 512, 256, 512, 512, 512, 1.0f, 1.0f);
    k_build_IpM<<<g1(512 * 512), blk, 0, stream>>>(G, Y3, IplusM, ImM);

    invert(IplusM, Ainv);
    // Ntop = (I - M) @ Ainv ; NbotRaw = Z3 @ Ainv
    gemm_f32_wmma<false, false><<<dim3(4, 4), blk, 0, stream>>>(
        ImM, Ainv, Ntop, 512, 512, 512, 512, 512, 512, 1.0f, 0.0f);
    gemm_f32_wmma<false, false><<<dim3(4, 2), blk, 0, stream>>>(
        Z3, Ainv, NbotR, 256, 512, 512, 512, 512, 512, 1.0f, 0.0f);
    k_build_D22<<<g1(PDIM * MDIM), blk, 0, stream>>>(Ntop, NbotR, D22);
    k_build_R1<<<g1(512 * 512), blk, 0, stream>>>(D22, R1);
    invert(R1, R1inv);

    k_build_Wmat<<<g1(NN2 * MDIM), blk, 0, stream>>>(Cm, Wu, Wmat);
    // T1 = Wmat @ R1inv ; H = Y @ Y^T + T1 @ Wmat^T + eps I
    gemm_f32_wmma<false, false><<<dim3(4, NN2 / 128), blk, 0, stream>>>(
        Wmat, R1inv, T1, NN2, 512, 512, 512, 512, 512, 1.0f, 0.0f);
    gemm_f32_wmma<false, true ><<<dim3(NN2 / 128, NN2 / 128), blk, 0, stream>>>(
        Ym, Ym, H, NN2, NN2, NN2, NN2, NN2, NN2, 1.0f, 0.0f);
    gemm_f32_wmma<false, true ><<<dim3(NN2 / 128, NN2 / 128), blk, 0, stream>>>(
        T1, Wmat, H, NN2, NN2, 512, 512, 512, NN2, 1.0f, 1.0f);
    k_H_add_eps<<<g1(NN2), blk, 0, stream>>>(H);

    // H21_star, Diff; delta = ||Diff||_2 + eps
    k_build_H21s<<<g1(LQ * NDIM), blk, 0, stream>>>(H, W1, H21s, Diff);
    specnorm(Diff, LQ, NDIM, NDIM, S_DELTA);

    // Pmat = H11 + delta I  -> straight into d_out
    k_build_Pmat<<<g1(NDIM * NDIM), blk, 0, stream>>>(H, scal, Pm);

    // P1 = H22 + delta I  (+ B^T Pmat B in last diagonal block)
    gemm_f32_wmma<true,  false><<<dim3(4, 4), blk, 0, stream>>>(
        Bm, Pm, BtP, 512, 512, 512, 512, 512, 512, 1.0f, 0.0f);
    gemm_f32_wmma<false, false><<<dim3(4, 4), blk, 0, stream>>>(
        BtP, Bm, BtPB, 512, 512, 512, 512, 512, 512, 1.0f, 0.0f);
    k_build_P1<<<g1(LQ * LQ), blk, 0, stream>>>(H, BtPB, scal, P1);

    // LambdaW2 blocks; residual (in place on P1); Ms = A^-1 residual
    k_build_LW2<<<g1(3 * QD * QD), blk, 0, stream>>>(P1, W1s, LW2);
    k_residual<<<g1(3 * QD * QD), blk, 0, stream>>>(P1, LW2);
    k_msolve<<<g1(LQ * LQ), blk, 0, stream>>>(P1, Ms);
    specnorm(Ms, LQ, LQ, LQ, S_LAMBDA);

    // effective weights
    k_alpha<<<g1(LQ), blk, 0, stream>>>(adp, scal, inva);
    k_wue<<<g1(QD * MDIM), blk, 0, stream>>>(Wu, inva, WuE);
    k_w1e<<<g1(LQ * NDIM), blk, 0, stream>>>(H21s, inva);     // H21s -> W1_eff
    k_w2e<<<g1(3 * QD * QD), blk, 0, stream>>>(LW2, inva, W2E);

    // ==== Scan ===========================================================
    // hoisted input projection: Upre = u @ Wu_eff0^T   (WMMA GEMM)
    gemm_f32_wmma<false, true ><<<dim3(4, TSTEP / 128), blk, 0, stream>>>(
        u, WuE, Upre, TSTEP, QD, MDIM, MDIM, MDIM, QD, 1.0f, 0.0f);

    k_recurrence<<<1, 1024, 0, stream>>>(Upre, H21s, W2E, Bm, bz, bx, x0,
                                         Xtm1, hout);

    // ==== Outputs ========================================================
    // Y_out = Xtm1 @ C^T + u @ D22^T + by
    gemm_f32_wmma<false, true ><<<dim3(PDIM / 128, TSTEP / 128), blk, 0, stream>>>(
        Xtm1, Cm, Yout, TSTEP, PDIM, NDIM, NDIM, NDIM, PDIM, 1.0f, 0.0f);
    gemm_f32_wmma<false, true ><<<dim3(PDIM / 128, TSTEP / 128), blk, 0, stream>>>(
        u, D22, Yout, TSTEP, PDIM, MDIM, MDIM, MDIM, PDIM, 1.0f, 1.0f);
    k_biasY<<<g1(TSTEP * PDIM), blk, 0, stream>>>(Yout, by);
}